// MultiDegradationRobustLoss_29334626631825
// MI455X (gfx1250) — compile-verified
//
#include <hip/hip_runtime.h>
#include <hip/hip_bf16.h>
#include <math.h>

typedef __attribute__((ext_vector_type(2))) float v2f;
typedef __attribute__((ext_vector_type(8))) float v8f;

#define NROW 8192
#define HALFB 4096
#define FDIM 128
#define LDA 132   // padded LDS row stride (floats): banks = 4*row + c, conflict-free
// 1/T and log2(e)/T for T = 0.07
#define INV_T 14.285714285714286f
#define LOG2E_OVER_T 20.609929155556620f

// Pin 8 v2f values into live VGPR pairs: forces the scheduler to keep the
// whole B tile resident instead of sinking each load into its WMMA use.
#define PIN8(x0,x1,x2,x3,x4,x5,x6,x7)                                \
    asm volatile("" : "+v"(x0), "+v"(x1), "+v"(x2), "+v"(x3),        \
                      "+v"(x4), "+v"(x5), "+v"(x6), "+v"(x7))

// ---------------- Kernel 1: per-row inverse L2 norm -------------------------
__global__ void norm_kernel(const float* __restrict__ feat,
                            float* __restrict__ scale) {
    const int row  = blockIdx.x * 8 + (threadIdx.x >> 5);
    const int lane = threadIdx.x & 31;
    const float4 v = *(const float4*)(feat + (size_t)row * FDIM + lane * 4);
    float ss = v.x * v.x + v.y * v.y + v.z * v.z + v.w * v.w;
#pragma unroll
    for (int m = 16; m >= 1; m >>= 1) ss += __shfl_xor(ss, m, 32);
    if (lane == 0) scale[row] = 1.0f / fmaxf(sqrtf(ss), 1e-12f);
}

// ---------------- Kernel 2: fused sim GEMM + masked logsumexp ---------------
// Block = 256 threads (8 waves). Block b owns rows [16b, 16b+16); the A tile
// (16x128) is staged once into padded LDS and hoisted to resident registers.
// Each wave sweeps 64 column tiles; the 32 B fragments of a tile are loaded
// as one clause, pinned live via inline asm, then consumed by an
// uninterrupted 32-deep WMMA chain.
__global__ void __launch_bounds__(256, 1)
simloss_kernel(const float* __restrict__ feat,
               const float* __restrict__ scale,
               float* __restrict__ lossRow) {
    __shared__ float lA[16 * LDA];
    __shared__ float sRow[16];
    __shared__ float sPos[16];

    const int tid  = threadIdx.x;
    const int lane = tid & 31;
    const int wave = tid >> 5;
    const int half = lane >> 4;   // 0: lanes 0-15, 1: lanes 16-31
    const int lr   = lane & 15;   // M (for A) / N (for B) index within tile
    const int m    = blockIdx.x;  // M tile index
    const int mBase = m * 16;

    if (tid < 16) { sRow[tid] = 0.0f; sPos[tid] = 0.0f; }

    // Cooperative stage of the A tile: thread t loads row t/16, 8 cols.
    {
        const int r = tid >> 4;
        const int c = (tid & 15) * 8;
        const float* src = feat + (size_t)(mBase + r) * FDIM + c;
        const float4 x0 = *(const float4*)(src);
        const float4 x1 = *(const float4*)(src + 4);
        float* dst = &lA[r * LDA + c];
        dst[0] = x0.x; dst[1] = x0.y; dst[2] = x0.z; dst[3] = x0.w;
        dst[4] = x1.x; dst[5] = x1.y; dst[6] = x1.z; dst[7] = x1.w;
    }
    __syncthreads();

    // Per-lane base of A fragments in LDS (V_WMMA_F32_16X16X4_F32 A layout:
    // lane lr holds row lr; step s covers K = 4s + 2*half + {0,1}).
    const float* aptr = &lA[lr * LDA + 2 * half];

    // Hoist all 32 A fragments to registers (loop-invariant across tiles).
    v2f a[32];
#pragma unroll
    for (int s = 0; s < 32; ++s)
        a[s] = *(const v2f*)(aptr + 4 * s);
    PIN8(a[0], a[1], a[2],  a[3],  a[4],  a[5],  a[6],  a[7]);
    PIN8(a[8], a[9], a[10], a[11], a[12], a[13], a[14], a[15]);
    PIN8(a[16],a[17],a[18], a[19], a[20], a[21], a[22], a[23]);
    PIN8(a[24],a[25],a[26], a[27], a[28], a[29], a[30], a[31]);

    // Row scales for the 8 C rows this lane holds (rows r + 8*half).
    float sA[8];
#pragma unroll
    for (int r = 0; r < 8; ++r) sA[r] = scale[mBase + r + 8 * half];

    float acc[8];
#pragma unroll
    for (int r = 0; r < 8; ++r) acc[r] = 0.0f;
    float posAcc = 0.0f;

    const int tBegin = wave * 64;
    const int tEnd   = tBegin + 64;
    for (int t = tBegin; t < tEnd; ++t) {
        // Load all 32 B fragments as a clause, then pin them live so the
        // WMMA chain below runs without per-instruction memory waits.
        const float* brow = feat + (size_t)(t * 16 + lr) * FDIM + 2 * half;
        v2f b[32];
#pragma unroll
        for (int s = 0; s < 32; ++s)
            b[s] = *(const v2f*)(brow + 4 * s);
        PIN8(b[0], b[1], b[2],  b[3],  b[4],  b[5],  b[6],  b[7]);
        PIN8(b[8], b[9], b[10], b[11], b[12], b[13], b[14], b[15]);
        PIN8(b[16],b[17],b[18], b[19], b[20], b[21], b[22], b[23]);
        PIN8(b[24],b[25],b[26], b[27], b[28], b[29], b[30], b[31]);

        v8f c0 = {0.f,0.f,0.f,0.f,0.f,0.f,0.f,0.f};
        v8f c1 = {0.f,0.f,0.f,0.f,0.f,0.f,0.f,0.f};
#pragma unroll
        for (int s = 0; s < 32; s += 2) {
            c0 = __builtin_amdgcn_wmma_f32_16x16x4_f32(
                     false, a[s],     false, b[s],     (short)0, c0, false, false);
            c1 = __builtin_amdgcn_wmma_f32_16x16x4_f32(
                     false, a[s + 1], false, b[s + 1], (short)0, c1, false, false);
        }

        // Epilogue: C element (r): row = mBase + r + 8*half, col = t*16 + lr.
        const int colg = t * 16 + lr;
        const float sB = scale[colg];
#pragma unroll
        for (int r = 0; r < 8; ++r) {
            const int rowg = mBase + r + 8 * half;
            const float val = (c0[r] + c1[r]) * sA[r] * sB;
            const float e = exp2f((val - 1.0f) * LOG2E_OVER_T);
            acc[r] += (colg == rowg) ? 0.0f : e;                     // skip diag
            posAcc += (colg == ((rowg + HALFB) & (NROW - 1))) ? val : 0.0f;
        }
    }

    // Cross-lane / cross-wave reduction of row sums via LDS float atomics.
#pragma unroll
    for (int r = 0; r < 8; ++r)
        atomicAdd(&sRow[r + 8 * half], acc[r]);
    // Positive (partner) terms live only on lanes 0-7 (row=lane) and
    // lanes 24-31 (row=lane-16).
    if (lane < 8)        atomicAdd(&sPos[lane], posAcc);
    else if (lane >= 24) atomicAdd(&sPos[lane - 16], posAcc);
    __syncthreads();

    if (tid < 16) {
        const float S = sRow[tid];               // sum_{j != i} exp((s_ij-1)/T)
        lossRow[mBase + tid] = INV_T + logf(S) - sPos[tid] * INV_T;
    }
}

// ---------------- Kernel 3: deterministic mean reduction --------------------
__global__ void reduce_kernel(const float* __restrict__ lossRow,
                              float* __restrict__ out) {
    __shared__ float buf[256];
    float s = 0.0f;
    for (int k = threadIdx.x; k < NROW; k += 256) s += lossRow[k];
    buf[threadIdx.x] = s;
    __syncthreads();
    for (int off = 128; off > 0; off >>= 1) {
        if (threadIdx.x < off) buf[threadIdx.x] += buf[threadIdx.x + off];
        __syncthreads();
    }
    if (threadIdx.x == 0) out[0] = buf[0] * (1.0f / (float)NROW);
}

extern "C" void kernel_launch(void* const* d_in, const int* in_sizes, int n_in,
                              void* d_out, int out_size, void* d_ws, size_t ws_size,
                              hipStream_t stream) {
    (void)in_sizes; (void)n_in; (void)out_size; (void)ws_size;
    const float* feat = (const float*)d_in[0];
    float* out     = (float*)d_out;
    float* scale   = (float*)d_ws;          // 8192 floats
    float* lossRow = scale + NROW;          // 8192 floats

    norm_kernel<<<NROW / 8, 256, 0, stream>>>(feat, scale);
    simloss_kernel<<<NROW / 16, 256, 0, stream>>>(feat, scale, lossRow);
    reduce_kernel<<<1, 256, 0, stream>>>(lossRow, out);
}